// MultiboxLoss_54563264529055
// MI455X (gfx1250) — compile-verified
//
#include <hip/hip_runtime.h>
#include <hip/hip_bf16.h>
#include <stdint.h>
#include <stddef.h>

// ---------------- problem constants ----------------
#define NUMC 21
#define BB   128
#define NN   8732
#define BN   (BB * NN)            // 1,117,696  (== 4366 * 256 exactly)
#define ROWS 256
#define NBLK1 (BN / ROWS)         // 4366
#define LOG_EPS (-16.118095650958319f)   // ln(1e-7)

typedef unsigned int uint32;
typedef uint32 v4u __attribute__((ext_vector_type(4)));
typedef int    v4i __attribute__((ext_vector_type(4)));
typedef int    v8i __attribute__((ext_vector_type(8)));

// ---------------- CDNA5 Tensor Data Mover path ----------------
#if defined(__AMDGCN__) && __has_builtin(__builtin_amdgcn_tensor_load_to_lds)
#define USE_TDM 1
#else
#define USE_TDM 0
#endif

#if USE_TDM
// Build a 1-D contiguous-tile TDM descriptor (D#) per CDNA5 ISA ch.10.7/§8:
//   group0: count=1 | lds_addr | global_addr(57b) | type=2
//   group1: data_size=4B, tensor_dim0 = #elements remaining (OOB -> zero-fill),
//           tensor_dim1=1, tile_dim0 = tile elements, tile_dim1/2 = 0 (unused)
__device__ __forceinline__ void tdm_load_1d(const void* gptr, void* ldsptr,
                                            uint32 tile_elems,
                                            unsigned long long tensor_elems)
{
  unsigned long long ga = (unsigned long long)(uintptr_t)gptr;
  uint32 lds = (uint32)(uintptr_t)ldsptr;   // flat LDS addr: low 32 bits = LDS byte offset
  uint32 td0 = (tensor_elems > 0xFFFFFFFFull) ? 0xFFFFFFFFu : (uint32)tensor_elems;

  v4u g0;
  g0.x = 1u;                                             // count=1 (valid), user mode
  g0.y = lds;                                            // lds_addr
  g0.z = (uint32)ga;                                     // global_addr[31:0]
  g0.w = (uint32)((ga >> 32) & 0x1FFFFFFu) | (2u << 30); // global_addr[56:32], type=2

  v8i g1;
  g1[0] = (int)(2u << 16);                       // data_size = 2 (4 bytes); no multicast/pad
  g1[1] = (int)((td0 & 0xFFFFu) << 16);          // tensor_dim0[15:0] @ bits 63:48
  g1[2] = (int)((td0 >> 16) | (1u << 16));       // tensor_dim0[31:16]; tensor_dim1 = 1
  g1[3] = (int)((tile_elems & 0xFFFFu) << 16);   // tensor_dim1 hi = 0; tile_dim0
  g1[4] = 0;                                     // tile_dim1 = 0, tile_dim2 = 0 (1-D)
  g1[5] = (int)td0;                              // tensor_dim0_stride lo (unused for 1-D)
  g1[6] = 0;
  g1[7] = 0;

  v4i z4 = {0, 0, 0, 0};
#if defined(__clang_major__) && (__clang_major__ >= 23)
  v8i z8 = {0, 0, 0, 0, 0, 0, 0, 0};
  __builtin_amdgcn_tensor_load_to_lds(g0, g1, z4, z4, z8, 0);
#else
  __builtin_amdgcn_tensor_load_to_lds(g0, g1, z4, z4, 0);
#endif
}
#endif // USE_TDM

// ---------------- K1: fused per-row math + key/loss emit ----------------
// ws layout handled by host side; scal[] = {posConf, posLoc, neg, denom, k, b1,
//                                           above1, P24, m, ticket, T32, m2, ...}
__global__ void __launch_bounds__(ROWS)
k_fuse(const float* __restrict__ yt, const float* __restrict__ lp,
       const float* __restrict__ cp, float* __restrict__ keys,
       float* __restrict__ closs, float* __restrict__ numpos,
       float* __restrict__ scal)
{
  __shared__ float s_conf[ROWS * NUMC];  // 21504 B
  __shared__ float s_yt[ROWS * 26];      // 26624 B
  __shared__ float s_loc[ROWS * 4];      //  4096 B
  const int t = threadIdx.x;
  const long long base = (long long)blockIdx.x * ROWS;

#if USE_TDM
  if (t == 0) {
    tdm_load_1d(cp + base * NUMC, s_conf, ROWS * NUMC,
                (unsigned long long)BN * NUMC - (unsigned long long)base * NUMC);
    tdm_load_1d(yt + base * 26, s_yt, ROWS * 26,
                (unsigned long long)BN * 26 - (unsigned long long)base * 26);
    tdm_load_1d(lp + base * 4, s_loc, ROWS * 4,
                (unsigned long long)BN * 4 - (unsigned long long)base * 4);
#if __has_builtin(__builtin_amdgcn_s_wait_tensorcnt)
    __builtin_amdgcn_s_wait_tensorcnt(0);   // wave-wide wait on TENSORcnt
#endif
  }
  __syncthreads();
#else
  {
    const float* gc = cp + base * NUMC;
    const float* gy = yt + base * 26;
    const float* gl = lp + base * 4;
    for (int i = t; i < ROWS * NUMC; i += ROWS) s_conf[i] = gc[i];
    for (int i = t; i < ROWS * 26;   i += ROWS) s_yt[i]   = gy[i];
    for (int i = t; i < ROWS * 4;    i += ROWS) s_loc[i]  = gl[i];
  }
  __syncthreads();
#endif

  const float* cf = &s_conf[t * NUMC];
  const float* y  = &s_yt[t * 26];
  const float* l  = &s_loc[t * 4];

  // numerically-stable softmax stats
  float mx = cf[0];
#pragma unroll
  for (int j = 1; j < NUMC; j++) mx = fmaxf(mx, cf[j]);
  float e0 = expf(cf[0] - mx);
  float ssum = e0, s1 = 0.f;
#pragma unroll
  for (int j = 1; j < NUMC; j++) { float e = expf(cf[j] - mx); ssum += e; s1 += e; }
  float lse = logf(ssum) + mx;

  // conf_loss = -sum_j onehot_j * log(max(softmax_j, 1e-7))
  float dot = 0.f;
#pragma unroll
  for (int j = 0; j < NUMC; j++) dot += y[4 + j] * fmaxf(cf[j] - lse, LOG_EPS);
  float cl = -dot;

  // smooth L1
  float ll = 0.f;
#pragma unroll
  for (int j = 0; j < 4; j++) {
    float d = y[j] - l[j];
    float ad = fabsf(d);
    ll += (ad < 1.f) ? 0.5f * d * d : (ad - 0.5f);
  }

  float pos = y[25];
  float key = (1.f - pos) * (s1 / ssum);   // (1-pos) * sum(softmax[1:])

  long long row = base + t;
  keys[row]  = key;
  closs[row] = cl;

  if (pos > 0.f) {   // ~2% of rows -> cheap sparse atomics
    int b = (int)(row / NN);
    atomicAdd(&numpos[b], pos);
    atomicAdd(&scal[0], cl * pos);
    atomicAdd(&scal[1], ll * pos);
  }
}

// ---------------- K2: per-batch scalars -> k_neg, denominator ----------------
__global__ void k_scalars(const float* __restrict__ numpos, float* __restrict__ scalf)
{
  __shared__ float rnn[128], rcl[128], rpos[128];
  int b = threadIdx.x;
  float np = numpos[b];
  rnn[b]  = fminf(3.0f * np, (float)NN - np);   // num_neg per batch
  rcl[b]  = (np != 0.f) ? np : 1.f;             // clamped num_pos
  rpos[b] = (np > 0.f) ? 1.f : 0.f;
  __syncthreads();
  for (int s = 64; s > 0; s >>= 1) {
    if (b < s) { rnn[b] += rnn[b + s]; rcl[b] += rcl[b + s]; rpos[b] += rpos[b + s]; }
    __syncthreads();
  }
  if (b == 0) {
    float k = (rpos[0] > 0.f) ? floorf(rnn[0]) : 100.0f;  // NEG_FOR_HARD
    ((uint32*)scalf)[4] = (uint32)k;
    scalf[3] = rcl[0];
  }
}

// ---------------- radix-select level 1: 12-bit histogram ----------------
__global__ void __launch_bounds__(256)
k_hist1(const float* __restrict__ keys, uint32* __restrict__ hist1)
{
  __shared__ uint32 lh[4096];
  for (int i = threadIdx.x; i < 4096; i += 256) lh[i] = 0;
  __syncthreads();
  long long stride = (long long)gridDim.x * blockDim.x;
  for (long long i = (long long)blockIdx.x * blockDim.x + threadIdx.x; i < BN; i += stride)
    atomicAdd(&lh[__float_as_uint(keys[i]) >> 20], 1u);
  __syncthreads();
  for (int i = threadIdx.x; i < 4096; i += 256)
    if (lh[i]) atomicAdd(&hist1[i], lh[i]);
}

__global__ void k_scan1(const uint32* __restrict__ hist1, uint32* __restrict__ scalu)
{
  uint32 k = scalu[4];
  if (k == 0u) { scalu[5] = 0xFFFFFFFFu; scalu[6] = 0u; return; }
  uint32 cum = 0, b1 = 0, above = 0; bool found = false;
  for (int b = 4095; b >= 0; b--) {
    uint32 c = hist1[b];
    if (cum + c >= k) { b1 = (uint32)b; above = cum; found = true; break; }
    cum += c;
  }
  if (!found) { b1 = 0; above = 0; }
  scalu[5] = b1; scalu[6] = above;
}

// ---------------- level 2: next 12 bits, only threshold bucket ----------------
__global__ void __launch_bounds__(256)
k_hist2(const float* __restrict__ keys, const uint32* __restrict__ scalu,
        uint32* __restrict__ hist2)
{
  uint32 b1 = scalu[5];
  long long stride = (long long)gridDim.x * blockDim.x;
  for (long long i = (long long)blockIdx.x * blockDim.x + threadIdx.x; i < BN; i += stride) {
    uint32 ku = __float_as_uint(keys[i]);
    if ((ku >> 20) == b1) atomicAdd(&hist2[(ku >> 8) & 0xFFFu], 1u);
  }
}

__global__ void k_scan2(const uint32* __restrict__ hist2, uint32* __restrict__ scalu)
{
  uint32 k = scalu[4];
  uint32 b1 = scalu[5];
  if (k == 0u || b1 == 0xFFFFFFFFu) { scalu[7] = 0xFFFFFFFFu; scalu[8] = 0u; return; }
  uint32 cum = scalu[6];
  uint32 P = 0, m = 0x7FFFFFFFu; bool found = false;
  for (int b = 4095; b >= 0; b--) {
    uint32 c = hist2[b];
    if (cum + c >= k) { P = (b1 << 12) | (uint32)b; m = k - cum; found = true; break; }
    cum += c;
  }
  if (!found) { P = 0; m = 0x7FFFFFFFu; }
  scalu[7] = P; scalu[8] = m;
}

// ---------------- level 3: last 8 bits -> exact 32-bit threshold ----------------
__global__ void __launch_bounds__(256)
k_hist3(const float* __restrict__ keys, const uint32* __restrict__ scalu,
        uint32* __restrict__ hist3)
{
  uint32 P = scalu[7];
  long long stride = (long long)gridDim.x * blockDim.x;
  for (long long i = (long long)blockIdx.x * blockDim.x + threadIdx.x; i < BN; i += stride) {
    uint32 ku = __float_as_uint(keys[i]);
    if ((ku >> 8) == P) atomicAdd(&hist3[ku & 0xFFu], 1u);
  }
}

__global__ void k_scan3(const uint32* __restrict__ hist3, uint32* __restrict__ scalu)
{
  uint32 P = scalu[7];
  uint32 m = scalu[8];
  if (P == 0xFFFFFFFFu) { scalu[10] = 0xFFFFFFFFu; scalu[11] = 0u; return; }
  uint32 cum = 0; uint32 T = 0, m2 = 0x7FFFFFFFu; bool found = false;
  for (int b = 255; b >= 0; b--) {
    uint32 c = hist3[b];
    if (cum + c >= m) { T = (P << 8) | (uint32)b; m2 = m - cum; found = true; break; }
    cum += c;
  }
  if (!found) { T = 0; m2 = 0x7FFFFFFFu; }
  scalu[10] = T; scalu[11] = m2;
}

// ---------------- K7: sum conf_loss over selected negatives ----------------
__global__ void __launch_bounds__(256)
k_negsum(const float* __restrict__ keys, const float* __restrict__ closs,
         float* __restrict__ scalf, uint32* __restrict__ scalu)
{
  uint32 T = scalu[10];
  uint32 m2 = scalu[11];
  float local = 0.f;
  long long stride = (long long)gridDim.x * blockDim.x;
  for (long long i = (long long)blockIdx.x * blockDim.x + threadIdx.x; i < BN; i += stride) {
    uint32 ku = __float_as_uint(keys[i]);
    if (ku > T) local += closs[i];
    else if (ku == T) {                      // exact-tie residue (measure-zero here)
      uint32 tk = atomicAdd(&scalu[9], 1u);
      if (tk < m2) local += closs[i];
    }
  }
  __shared__ float red[256];
  red[threadIdx.x] = local;
  __syncthreads();
  for (int s = 128; s > 0; s >>= 1) {
    if (threadIdx.x < s) red[threadIdx.x] += red[threadIdx.x + s];
    __syncthreads();
  }
  if (threadIdx.x == 0 && red[0] != 0.f) atomicAdd(&scalf[2], red[0]);
}

// ---------------- K8: finalize ----------------
__global__ void k_final(const float* __restrict__ scalf, float* __restrict__ out)
{
  out[0] = (scalf[0] + scalf[2] + 1.0f * scalf[1]) / scalf[3];   // ALPHA = 1
}

// ---------------- host side ----------------
extern "C" void kernel_launch(void* const* d_in, const int* in_sizes, int n_in,
                              void* d_out, int out_size, void* d_ws, size_t ws_size,
                              hipStream_t stream)
{
  (void)in_sizes; (void)n_in; (void)out_size; (void)ws_size;
  const float* yt = (const float*)d_in[0];   // y_true   (B,N,26)
  const float* lp = (const float*)d_in[1];   // loc_pred (B,N,4)
  const float* cp = (const float*)d_in[2];   // conf_pred(B,N,21)

  float*  keys   = (float*)d_ws;
  float*  closs  = keys + BN;
  uint32* hist1  = (uint32*)(closs + BN);
  uint32* hist2  = hist1 + 4096;
  uint32* hist3  = hist2 + 4096;
  float*  numpos = (float*)(hist3 + 256);
  float*  scalf  = numpos + 128;
  uint32* scalu  = (uint32*)scalf;

  // zero all accumulators/histograms/scalars (contiguous region) every call
  hipMemsetAsync(hist1, 0, (4096 + 4096 + 256 + 128 + 16) * sizeof(uint32), stream);

  k_fuse   <<<NBLK1, ROWS, 0, stream>>>(yt, lp, cp, keys, closs, numpos, scalf);
  k_scalars<<<1, 128, 0, stream>>>(numpos, scalf);
  k_hist1  <<<512, 256, 0, stream>>>(keys, hist1);
  k_scan1  <<<1, 1, 0, stream>>>(hist1, scalu);
  k_hist2  <<<512, 256, 0, stream>>>(keys, scalu, hist2);
  k_scan2  <<<1, 1, 0, stream>>>(hist2, scalu);
  k_hist3  <<<512, 256, 0, stream>>>(keys, scalu, hist3);
  k_scan3  <<<1, 1, 0, stream>>>(hist3, scalu);
  k_negsum <<<512, 256, 0, stream>>>(keys, closs, scalf, scalu);
  k_final  <<<1, 1, 0, stream>>>(scalf, (float*)d_out);
}